// GNNLSTM_51908974740047
// MI455X (gfx1250) — compile-verified
//
#include <hip/hip_runtime.h>
#include <hip/hip_bf16.h>

// ---------------------------------------------------------------------------
// GCN(2) + LSTM(2) + FC for MI455X (gfx1250, wave32).
//
// GEMMs run on V_WMMA_F32_16X16X4_F32 (full fp32; the workload is ~30 GFLOP
// and memory/latency bound, so low-precision WMMA buys nothing).
// Round-2 change: each wave register-caches the weight (B) fragments for its
// 16-column strip across the whole K=128 reduction (32 x v2f = 64 VGPRs per
// matrix), then sweeps multiple M-tiles reusing them. The fully unrolled
// K-loop issues 32/64 back-to-back v_wmma per tile with only contiguous
// b64 A-loads in between -> far fewer VMEM ops per WMMA.
// Edge scatter uses fp32 global atomics; h/agg (25.6 MB each) are L2-resident
// in the 192 MB L2.
// ---------------------------------------------------------------------------

#define N_NODES 50000
#define N_EDGES 600000
#define DIM     128          // feature dim == hidden dim == K
#define SEQL    50
#define BATCH   1000         // N_NODES / SEQL
#define GDIM    512          // 4*H gate width
#define KSTEPS  (DIM / 4)    // 32 WMMA k-steps over K=128

typedef float v2f __attribute__((ext_vector_type(2)));
typedef float v8f __attribute__((ext_vector_type(8)));

// ---------------------------------------------------------------------------
// Fragment layouts for V_WMMA_F32_16X16X4_F32 (wave32), per ISA:
//   A 16x4 : lane L holds row m=L&15; VGPRs = K {0,1} (lanes 0-15) or
//            K {2,3} (lanes 16-31)  -> contiguous float2 at k + 2*(L>=16)
//   B 4x16 : lane L holds col n=L&15; same K split
//   C/D    : VGPR r -> row r (lanes 0-15) / row r+8 (lanes 16-31), col = L&15
// ---------------------------------------------------------------------------

__device__ __forceinline__ void store_tile_f32(float* __restrict__ C, int ldc,
                                               int rowBase, int colBase, int M, v8f c)
{
    const int lane = threadIdx.x & 31;
    const int rhalf = (lane >> 4) << 3;     // +0 or +8 row offset
    const int n = lane & 15;
#pragma unroll
    for (int r = 0; r < 8; ++r) {
        int row = rowBase + r + rhalf;
        if (row < M) C[(size_t)row * ldc + colBase + n] = c[r];
    }
}

// ---------------------------------------------------------------------------
// GCN feature transform: C(MxN) = A(Mx128) @ B(128xN), both row-major.
// One wave owns one 16-wide N strip; caches the full B strip in registers
// (32 x v2f), then sweeps MT consecutive M-tiles.
// ---------------------------------------------------------------------------
#define MT_GCN 4
__global__ void gcn_gemm_wmma(const float* __restrict__ A,
                              const float* __restrict__ B,
                              float* __restrict__ C, int M, int N)
{
    const int tilesM  = (M + 15) >> 4;
    const int mGroups = (tilesM + MT_GCN - 1) / MT_GCN;
    const int waveId  = blockIdx.x * blockDim.y + threadIdx.y;
    if (waveId >= mGroups) return;          // wave-uniform
    const int tN   = blockIdx.y;
    const int lane = threadIdx.x & 31;
    const int khalf = (lane >> 4) << 1;     // 0 | 2
    const int mn   = lane & 15;
    const int colBase = tN * 16;

    // register-cache the B strip: B[k+khalf][colBase+mn] for k = 0,4,...,124
    v2f bfrag[KSTEPS];
#pragma unroll
    for (int kk = 0; kk < KSTEPS; ++kk) {
        const float* bp = B + (size_t)(kk * 4 + khalf) * N + colBase + mn;
        bfrag[kk].x = bp[0];
        bfrag[kk].y = bp[N];
    }

    for (int mt = 0; mt < MT_GCN; ++mt) {
        const int tM = waveId * MT_GCN + mt;
        if (tM >= tilesM) break;            // wave-uniform
        int arow = tM * 16 + mn;
        if (arow >= M) arow = M - 1;        // clamp reads; stores guarded
        const float* aptr = A + (size_t)arow * DIM + khalf;
        v8f acc = {};
#pragma unroll
        for (int kk = 0; kk < KSTEPS; ++kk) {
            v2f a;                          // contiguous -> global_load_b64
            a.x = aptr[kk * 4];
            a.y = aptr[kk * 4 + 1];
            // (neg_a, A, neg_b, B, c_mod, C, reuse_a, reuse_b)
            acc = __builtin_amdgcn_wmma_f32_16x16x4_f32(false, a, false, bfrag[kk],
                                                        (short)0, acc, false, false);
        }
        store_tile_f32(C, N, tM * 16, colBase, M, acc);
    }
}

// ---------------------------------------------------------------------------
// LSTM gate pre-activations for one timestep:
//   G(1000x512) = Xt(1000x128) @ Wih^T + Hprev(1000x128) @ Whh^T + (bih+bhh)
// Wih/Whh are (512x128) row-major, used transposed: B[k][n] = W[n*128+k]
// -> contiguous float2 fragments. Both weight strips cached in registers
// (128 VGPRs), wave sweeps MT_LSTM M-tiles.
// ---------------------------------------------------------------------------
#define MT_LSTM 2
__global__ void lstm_gates_wmma(const float* __restrict__ Xt, int ldx,
                                const float* __restrict__ Hprev,
                                const float* __restrict__ Wih,
                                const float* __restrict__ Whh,
                                const float* __restrict__ bih,
                                const float* __restrict__ bhh,
                                float* __restrict__ G)
{
    const int tilesM  = (BATCH + 15) >> 4;                   // 63
    const int mGroups = (tilesM + MT_LSTM - 1) / MT_LSTM;    // 32
    const int waveId  = blockIdx.x * blockDim.y + threadIdx.y;
    if (waveId >= mGroups) return;
    const int tN   = blockIdx.y;
    const int lane = threadIdx.x & 31;
    const int khalf = (lane >> 4) << 1;
    const int mn   = lane & 15;
    const int colBase = tN * 16;

    v2f bi[KSTEPS], bh[KSTEPS];
#pragma unroll
    for (int kk = 0; kk < KSTEPS; ++kk) {
        const float* p0 = Wih + (size_t)(colBase + mn) * DIM + kk * 4 + khalf;
        const float* p1 = Whh + (size_t)(colBase + mn) * DIM + kk * 4 + khalf;
        bi[kk].x = p0[0]; bi[kk].y = p0[1];
        bh[kk].x = p1[0]; bh[kk].y = p1[1];
    }
    const int col = colBase + mn;
    const float bias = bih[col] + bhh[col];

    for (int mt = 0; mt < MT_LSTM; ++mt) {
        const int tM = waveId * MT_LSTM + mt;
        if (tM >= tilesM) break;            // wave-uniform
        int arow = tM * 16 + mn;
        if (arow >= BATCH) arow = BATCH - 1;
        const float* xp = Xt    + (size_t)arow * ldx + khalf;
        const float* hp = Hprev + (size_t)arow * DIM + khalf;
        v8f acc = {};
#pragma unroll
        for (int kk = 0; kk < KSTEPS; ++kk) {
            v2f a; a.x = xp[kk * 4]; a.y = xp[kk * 4 + 1];
            acc = __builtin_amdgcn_wmma_f32_16x16x4_f32(false, a, false, bi[kk],
                                                        (short)0, acc, false, false);
        }
#pragma unroll
        for (int kk = 0; kk < KSTEPS; ++kk) {
            v2f a; a.x = hp[kk * 4]; a.y = hp[kk * 4 + 1];
            acc = __builtin_amdgcn_wmma_f32_16x16x4_f32(false, a, false, bh[kk],
                                                        (short)0, acc, false, false);
        }
        const int rhalf = (lane >> 4) << 3;
#pragma unroll
        for (int r = 0; r < 8; ++r) {
            int row = tM * 16 + r + rhalf;
            if (row < BATCH) G[(size_t)row * GDIM + col] = acc[r] + bias;
        }
    }
}

// ---------------------------------------------------------------------------
// Graph pieces
// ---------------------------------------------------------------------------
__global__ void deg_count(const int* __restrict__ dst, float* __restrict__ deg)
{
    int e = blockIdx.x * blockDim.x + threadIdx.x;
    if (e < N_EDGES) atomicAdd(&deg[dst[e]], 1.0f);
}

__global__ void compute_dinv(float* __restrict__ d)   // in-place deg -> rsqrt(deg+1)
{
    int n = blockIdx.x * blockDim.x + threadIdx.x;
    if (n < N_NODES) d[n] = rsqrtf(d[n] + 1.0f);
}

// One wave per edge; each lane moves 4 floats (coalesced 512B row per wave).
__global__ void edge_scatter(const int* __restrict__ src, const int* __restrict__ dst,
                             const float* __restrict__ dinv,
                             const float* __restrict__ h, float* __restrict__ agg)
{
    size_t tid = (size_t)blockIdx.x * blockDim.x + threadIdx.x;
    if (tid >= (size_t)N_EDGES * 32) return;
    const int e = (int)(tid >> 5);
    const int lane = (int)(tid & 31);
    const int s = src[e], d = dst[e];
    const float nrm = dinv[s] * dinv[d];
    const float4 hv = *(const float4*)(h + (size_t)s * DIM + lane * 4);
    float* ap = agg + (size_t)d * DIM + lane * 4;
    atomicAdd(ap + 0, hv.x * nrm);
    atomicAdd(ap + 1, hv.y * nrm);
    atomicAdd(ap + 2, hv.z * nrm);
    atomicAdd(ap + 3, hv.w * nrm);
}

// agg = relu(agg + h*dinv^2 + b)   (self-loop term + bias + activation)
__global__ void gcn_finalize(const float* __restrict__ h, const float* __restrict__ dinv,
                             const float* __restrict__ bias, float* __restrict__ agg)
{
    size_t tid = (size_t)blockIdx.x * blockDim.x + threadIdx.x;
    if (tid >= (size_t)N_NODES * 32) return;
    const int n = (int)(tid >> 5);
    const int lane = (int)(tid & 31);
    const float di = dinv[n];
    const float d2 = di * di;
    const float4 hv = *(const float4*)(h + (size_t)n * DIM + lane * 4);
    const float4 bv = *(const float4*)(bias + lane * 4);
    float4* ap = (float4*)(agg + (size_t)n * DIM + lane * 4);
    float4 av = *ap;
    av.x = fmaxf(av.x + hv.x * d2 + bv.x, 0.0f);
    av.y = fmaxf(av.y + hv.y * d2 + bv.y, 0.0f);
    av.z = fmaxf(av.z + hv.z * d2 + bv.z, 0.0f);
    av.w = fmaxf(av.w + hv.w * d2 + bv.w, 0.0f);
    *ap = av;
}

__device__ __forceinline__ float sigf(float x) { return 1.0f / (1.0f + __expf(-x)); }

// Elementwise LSTM cell update; optionally writes h into the output sequence.
__global__ void lstm_cell(const float* __restrict__ G, float* __restrict__ h,
                          float* __restrict__ c, float* __restrict__ seq_out, int t)
{
    int idx = blockIdx.x * blockDim.x + threadIdx.x;
    if (idx >= BATCH * DIM) return;
    const int b = idx >> 7;
    const int j = idx & (DIM - 1);
    const float* g = G + (size_t)b * GDIM;
    const float ig = sigf(g[j]);
    const float fg = sigf(g[DIM + j]);
    const float gg = tanhf(g[2 * DIM + j]);
    const float og = sigf(g[3 * DIM + j]);
    const float cn = fg * c[idx] + ig * gg;
    const float hn = og * tanhf(cn);
    c[idx] = cn;
    h[idx] = hn;
    if (seq_out) seq_out[((size_t)b * SEQL + t) * DIM + j] = hn;
}

__global__ void fc_kernel(const float* __restrict__ h, const float* __restrict__ w,
                          const float* __restrict__ b, float* __restrict__ out)
{
    int i = blockIdx.x * blockDim.x + threadIdx.x;
    if (i >= BATCH) return;
    const float* hp = h + (size_t)i * DIM;
    float s = 0.0f;
#pragma unroll 4
    for (int j = 0; j < DIM; ++j) s += hp[j] * w[j];
    out[i] = s + b[0];
}

// ---------------------------------------------------------------------------
// Host driver — everything on `stream`, graph-capture safe.
// ---------------------------------------------------------------------------
extern "C" void kernel_launch(void* const* d_in, const int* in_sizes, int n_in,
                              void* d_out, int out_size, void* d_ws, size_t ws_size,
                              hipStream_t stream)
{
    const float* x    = (const float*)d_in[0];
    const int*  eidx  = (const int*)  d_in[1];
    const int*  esrc  = eidx;
    const int*  edst  = eidx + N_EDGES;
    const float* W1   = (const float*)d_in[3];
    const float* b1   = (const float*)d_in[4];
    const float* W2   = (const float*)d_in[5];
    const float* b2   = (const float*)d_in[6];
    const float* fcw  = (const float*)d_in[7];
    const float* fcb  = (const float*)d_in[8];
    const float* wih0 = (const float*)d_in[9];
    const float* whh0 = (const float*)d_in[10];
    const float* bih0 = (const float*)d_in[11];
    const float* bhh0 = (const float*)d_in[12];
    const float* wih1 = (const float*)d_in[13];
    const float* whh1 = (const float*)d_in[14];
    const float* bih1 = (const float*)d_in[15];
    const float* bhh1 = (const float*)d_in[16];
    float* out = (float*)d_out;

    // workspace layout (floats), ~80 MB total
    float* ws   = (float*)d_ws;
    float* dinv = ws;                                 // N
    float* bufA = dinv + N_NODES;                     // N*128 : GEMM output h
    float* bufB = bufA + (size_t)N_NODES * DIM;       // N*128 : layer1 out / seq1 out
    float* bufC = bufB + (size_t)N_NODES * DIM;       // N*128 : layer2 out (LSTM input)
    float* G    = bufC + (size_t)N_NODES * DIM;       // 1000*512 gate pre-activations
    float* hs   = G + (size_t)BATCH * GDIM;           // 1000*128 hidden state
    float* cs   = hs + (size_t)BATCH * DIM;           // 1000*128 cell state

    const dim3 wblk(32, 4);                           // 4 waves / block

    // GCN GEMM: tilesM = 3125, mGroups = 782 -> 196 blocks x; 8 N strips
    const int gcn_groups = ((N_NODES + 15) / 16 + MT_GCN - 1) / MT_GCN;
    const dim3 ggrid_gcn((gcn_groups + 3) / 4, DIM / 16);

    // LSTM GEMM: tilesM = 63, mGroups = 32 -> 8 blocks x; 32 N strips
    const int lstm_groups = ((BATCH + 15) / 16 + MT_LSTM - 1) / MT_LSTM;
    const dim3 ggrid_lstm((lstm_groups + 3) / 4, GDIM / 16);

    const int scat_blocks = (int)(((size_t)N_EDGES * 32 + 255) / 256);
    const int fin_blocks  = (int)(((size_t)N_NODES * 32 + 255) / 256);

    // degree -> dinv (edges fixed, compute once for both GCN layers)
    hipMemsetAsync(dinv, 0, N_NODES * sizeof(float), stream);
    deg_count<<<(N_EDGES + 255) / 256, 256, 0, stream>>>(edst, dinv);
    compute_dinv<<<(N_NODES + 255) / 256, 256, 0, stream>>>(dinv);

    // ---- GCN layer 1: bufB = relu(scatter(x@W1) + (x@W1)*dinv^2 + b1)
    gcn_gemm_wmma<<<ggrid_gcn, wblk, 0, stream>>>(x, W1, bufA, N_NODES, DIM);
    hipMemsetAsync(bufB, 0, (size_t)N_NODES * DIM * sizeof(float), stream);
    edge_scatter<<<scat_blocks, 256, 0, stream>>>(esrc, edst, dinv, bufA, bufB);
    gcn_finalize<<<fin_blocks, 256, 0, stream>>>(bufA, dinv, b1, bufB);

    // ---- GCN layer 2: bufC = relu(scatter(bufB@W2) + ... + b2)
    gcn_gemm_wmma<<<ggrid_gcn, wblk, 0, stream>>>(bufB, W2, bufA, N_NODES, DIM);
    hipMemsetAsync(bufC, 0, (size_t)N_NODES * DIM * sizeof(float), stream);
    edge_scatter<<<scat_blocks, 256, 0, stream>>>(esrc, edst, dinv, bufA, bufC);
    gcn_finalize<<<fin_blocks, 256, 0, stream>>>(bufA, dinv, b2, bufC);

    // ---- LSTM layer 1: input seq = bufC (row n = b*SEQL + t), output seq -> bufB
    hipMemsetAsync(hs, 0, (size_t)BATCH * DIM * sizeof(float), stream);
    hipMemsetAsync(cs, 0, (size_t)BATCH * DIM * sizeof(float), stream);
    for (int t = 0; t < SEQL; ++t) {
        lstm_gates_wmma<<<ggrid_lstm, wblk, 0, stream>>>(
            bufC + (size_t)t * DIM, SEQL * DIM, hs, wih0, whh0, bih0, bhh0, G);
        lstm_cell<<<(BATCH * DIM + 255) / 256, 256, 0, stream>>>(G, hs, cs, bufB, t);
    }

    // ---- LSTM layer 2: input seq = bufB; only final h needed
    hipMemsetAsync(hs, 0, (size_t)BATCH * DIM * sizeof(float), stream);
    hipMemsetAsync(cs, 0, (size_t)BATCH * DIM * sizeof(float), stream);
    for (int t = 0; t < SEQL; ++t) {
        lstm_gates_wmma<<<ggrid_lstm, wblk, 0, stream>>>(
            bufB + (size_t)t * DIM, SEQL * DIM, hs, wih1, whh1, bih1, bhh1, G);
        lstm_cell<<<(BATCH * DIM + 255) / 256, 256, 0, stream>>>(G, hs, cs, nullptr, t);
    }

    // ---- FC head: out(1000) = hs @ fc_w^T + fc_b
    fc_kernel<<<(BATCH + 255) / 256, 256, 0, stream>>>(hs, fcw, fcb, out);
}